// DihedralBiasVmap_9113920602375
// MI455X (gfx1250) — compile-verified
//
#include <hip/hip_runtime.h>
#include <math.h>

#define N_CVS 30000
#define NX 60000            // 2*N_CVS
#define N_MODELS 8
#define LN_EPS 1e-5f
#define PI_F 3.14159265358979323846f

typedef float v2f __attribute__((ext_vector_type(2)));
typedef float v8f __attribute__((ext_vector_type(8)));

// ---- workspace layout (float offsets) ----
#define WS_XRAW   0           // 60000: [cos(cv); sin(cv)]
#define WS_DCVS   60000       // 360000: d(cv)/dx, 12 per cv
#define WS_DXHAT  420000      // 480000: per-model dE/dxhat
#define WS_MF     900000      // 30000: mean force per cv
#define WS_Y0     930000      // 512: layer0 pre-activations
#define WS_G0     930512      // 512: grad at layer0 pre-act
#define WS_E      931024      // 8: per-model energy
#define WS_S1     931032      // 8
#define WS_S2     931040      // 8
#define WS_SUM    931048
#define WS_SUMSQ  931049
#define WS_SS     931050
#define WS_SIGMA  931051
#define WS_ACC_BEGIN 930000
#define WS_ACC_COUNT 1052     // divisible by 4

// forward GEMV partition: 8 models x 4 jtiles x FWD_CH chunks (1 wave each)
#define FWD_CH    375
#define FWD_STEPS 40          // 375*40*4 = 60000
// backward GEMV partition: 8 models x BWD_CH chunks (1 wave each)
#define BWD_CH    750
#define BWD_TILES 5           // 750*5*16 = 60000

__device__ __forceinline__ v8f wmma4(v2f a, v2f b, v8f c) {
  return __builtin_amdgcn_wmma_f32_16x16x4_f32(false, a, false, b, (short)0, c,
                                               false, false);
}

struct F3 { float x, y, z; };
__device__ __forceinline__ F3 fsub(F3 a, F3 b){ return {a.x-b.x, a.y-b.y, a.z-b.z}; }
__device__ __forceinline__ F3 fadd(F3 a, F3 b){ return {a.x+b.x, a.y+b.y, a.z+b.z}; }
__device__ __forceinline__ F3 fscl(F3 a, float s){ return {a.x*s, a.y*s, a.z*s}; }
__device__ __forceinline__ float fdot(F3 a, F3 b){ return a.x*b.x + a.y*b.y + a.z*b.z; }
__device__ __forceinline__ F3 fcross(F3 a, F3 b){
  return {a.y*b.z - a.z*b.y, a.z*b.x - a.x*b.z, a.x*b.y - a.y*b.x};
}

// ---------------- K0: zero helpers (B128 stores for the bulk) ----------------
__global__ void k_zero4(float4* __restrict__ p, int n4) {
  int i = blockIdx.x * blockDim.x + threadIdx.x;
  int stride = gridDim.x * blockDim.x;
  float4 z = {0.f, 0.f, 0.f, 0.f};
  for (; i < n4; i += stride) p[i] = z;
}
__global__ void k_zero1(float* __restrict__ p, int n) {
  int i = blockIdx.x * blockDim.x + threadIdx.x;
  if (i < n) p[i] = 0.f;
}

// ---------------- K1: dihedrals + analytic grads + LN stats ----------------
__global__ void k_dihedral(const float* __restrict__ pos,
                           const float* __restrict__ boxv,
                           const int* __restrict__ cidx,
                           float* __restrict__ ws) {
  __shared__ float r1[256], r2[256];
  int tid = threadIdx.x;
  int c = blockIdx.x * 256 + tid;
  float ls = 0.f, lq = 0.f;
  if (c < N_CVS) {
    float bx = boxv[0], by = boxv[4], bz = boxv[8];
    F3 p[4];
    #pragma unroll
    for (int a = 0; a < 4; ++a) {
      int pi = cidx[c * 4 + a];
      float x = pos[pi * 3 + 0], y = pos[pi * 3 + 1], z = pos[pi * 3 + 2];
      p[a] = {x - floorf(x / bx) * bx, y - floorf(y / by) * by,
              z - floorf(z / bz) * bz};
    }
    F3 b1 = fsub(p[1], p[0]), b2 = fsub(p[2], p[1]), b3 = fsub(p[3], p[2]);
    F3 n1 = fcross(b1, b2), n2 = fcross(b2, b3);
    float nb2sq = fdot(b2, b2);
    float nb2 = sqrtf(nb2sq);
    F3 m1 = fcross(n1, fscl(b2, 1.f / nb2));
    float ys = fdot(m1, n2), xc = fdot(n1, n2);
    float rinv = rsqrtf(xc * xc + ys * ys);
    float cph = xc * rinv, sph = ys * rinv;   // cos(phi), sin(phi)
    ws[WS_XRAW + c] = cph;
    ws[WS_XRAW + N_CVS + c] = sph;
    // gradient (reference's atan2(m1.n2, n1.n2) sign convention)
    float in1 = 1.f / fdot(n1, n1), in2 = 1.f / fdot(n2, n2);
    F3 g0 = fscl(n1, nb2 * in1);
    F3 g3 = fscl(n2, -nb2 * in2);
    float s12 = fdot(b1, b2) / nb2sq, s32 = fdot(b3, b2) / nb2sq;
    F3 g1 = fadd(fscl(g0, s12 - 1.f), fscl(g3, s32));
    F3 g2 = fscl(fadd(fadd(g0, g1), g3), -1.f);
    float* d = ws + WS_DCVS + (size_t)c * 12;
    d[0]=g0.x; d[1]=g0.y; d[2]=g0.z; d[3]=g1.x; d[4]=g1.y; d[5]=g1.z;
    d[6]=g2.x; d[7]=g2.y; d[8]=g2.z; d[9]=g3.x; d[10]=g3.y; d[11]=g3.z;
    ls = cph + sph;
    lq = cph * cph + sph * sph;
  }
  r1[tid] = ls; r2[tid] = lq;
  __syncthreads();
  for (int s = 128; s > 0; s >>= 1) {
    if (tid < s) { r1[tid] += r1[tid + s]; r2[tid] += r2[tid + s]; }
    __syncthreads();
  }
  if (tid == 0) {
    atomicAdd(&ws[WS_SUM], r1[0]);
    atomicAdd(&ws[WS_SUMSQ], r2[0]);
  }
}

// ---------------- K2: layer0 forward GEMV via f32 WMMA ----------------
// y0pre[m,j] = sum_i xn[i] * W0[m,i,j]; D[r,n] = sum_k A[r,k]B[k,n]
// A[r,k] = W0[m, i0+k, j0+r] (16 j x 4 i), B[k,n] = xn[i0+k] broadcast.
__global__ void k_layer0_fwd(const float* __restrict__ W0,
                             const float* __restrict__ gamma,
                             const float* __restrict__ beta,
                             float* __restrict__ ws) {
  const int wave = threadIdx.x >> 5;
  const int lane = threadIdx.x & 31;
  int unit = blockIdx.x * 8 + wave;        // (m*4 + jt)*FWD_CH + ch
  int ch = unit % FWD_CH;
  int mj = unit / FWD_CH;
  int jt = mj & 3;
  int m = mj >> 2;
  float mu = ws[WS_SUM] * (1.f / NX);
  float var = ws[WS_SUMSQ] * (1.f / NX) - mu * mu;
  float rs = rsqrtf(var + LN_EPS);
  int hi = lane >> 4, lo = lane & 15;
  int j0 = jt * 16;
  const float* Wm = W0 + (size_t)m * NX * 64;
  v8f acc = {0.f,0.f,0.f,0.f,0.f,0.f,0.f,0.f};
  int ibase = ch * (FWD_STEPS * 4);
  for (int s = 0; s < FWD_STEPS; ++s) {
    int i0 = ibase + s * 4;
    int ri = i0 + 2 * hi;                  // rows i0,i0+1 (half 0) / i0+2,i0+3 (half 1)
    v2f a, b;
    a.x = Wm[(size_t)ri * 64 + j0 + lo];
    a.y = Wm[(size_t)(ri + 1) * 64 + j0 + lo];
    float x0 = ws[WS_XRAW + ri], x1 = ws[WS_XRAW + ri + 1];
    b.x = (x0 - mu) * rs * gamma[ri] + beta[ri];
    b.y = (x1 - mu) * rs * gamma[ri + 1] + beta[ri + 1];
    acc = wmma4(a, b, acc);
  }
  float* y0 = ws + WS_Y0 + m * 64 + j0;
  if (lane == 0) {
    #pragma unroll
    for (int v = 0; v < 8; ++v) atomicAdd(&y0[v], acc[v]);       // rows 0..7
  }
  if (lane == 16) {
    #pragma unroll
    for (int v = 0; v < 8; ++v) atomicAdd(&y0[8 + v], acc[v]);   // rows 8..15
  }
}

// ---------------- K3: tiny MLP (layers 1..4) fwd + bwd, per model ----------------
__global__ void k_mlp(const float* __restrict__ b0, const float* __restrict__ W1,
                      const float* __restrict__ b1, const float* __restrict__ W2,
                      const float* __restrict__ b2, const float* __restrict__ W3,
                      const float* __restrict__ b3, const float* __restrict__ W4,
                      const float* __restrict__ b4, float* __restrict__ ws) {
  int m = blockIdx.x;
  int t = threadIdx.x;                     // 64 threads
  __shared__ float pre[4][64];
  __shared__ float g[64];
  __shared__ float red[64];
  const float* w1 = W1 + m * 4096;
  const float* w2 = W2 + m * 4096;
  const float* w3 = W3 + m * 4096;
  pre[0][t] = ws[WS_Y0 + m * 64 + t] + b0[m * 64 + t];
  __syncthreads();
  float acc = 0.f;
  for (int i = 0; i < 64; ++i) acc += fmaxf(pre[0][i], 0.f) * w1[i * 64 + t];
  pre[1][t] = acc + b1[m * 64 + t];
  __syncthreads();
  acc = 0.f;
  for (int i = 0; i < 64; ++i) acc += fmaxf(pre[1][i], 0.f) * w2[i * 64 + t];
  pre[2][t] = acc + b2[m * 64 + t];
  __syncthreads();
  acc = 0.f;
  for (int i = 0; i < 64; ++i) acc += fmaxf(pre[2][i], 0.f) * w3[i * 64 + t];
  pre[3][t] = acc + b3[m * 64 + t];
  __syncthreads();
  red[t] = fmaxf(pre[3][t], 0.f) * W4[m * 64 + t];
  __syncthreads();
  for (int s = 32; s > 0; s >>= 1) {
    if (t < s) red[t] += red[t + s];
    __syncthreads();
  }
  if (t == 0) ws[WS_E + m] = red[0] + b4[m];
  // backward
  g[t] = (pre[3][t] > 0.f) ? W4[m * 64 + t] : 0.f;   // g3
  __syncthreads();
  float a2 = 0.f;
  for (int j = 0; j < 64; ++j) a2 += w3[t * 64 + j] * g[j];
  a2 = (pre[2][t] > 0.f) ? a2 : 0.f;
  __syncthreads(); g[t] = a2; __syncthreads();
  float a1 = 0.f;
  for (int j = 0; j < 64; ++j) a1 += w2[t * 64 + j] * g[j];
  a1 = (pre[1][t] > 0.f) ? a1 : 0.f;
  __syncthreads(); g[t] = a1; __syncthreads();
  float a0 = 0.f;
  for (int j = 0; j < 64; ++j) a0 += w1[t * 64 + j] * g[j];
  a0 = (pre[0][t] > 0.f) ? a0 : 0.f;
  ws[WS_G0 + m * 64 + t] = a0;
}

// ---------------- K4: layer0 backward GEMV via f32 WMMA (L2-resident pass) ----
// dxn[m,i] = sum_j W0[m,i,j] * g0[m,j]; A[r,k]=W0[m,i0+r,j0+k], B[k,n]=g0 bcast.
__global__ void k_layer0_bwd(const float* __restrict__ W0,
                             const float* __restrict__ gamma,
                             float* __restrict__ ws) {
  const int wave = threadIdx.x >> 5;
  const int lane = threadIdx.x & 31;
  int unit = blockIdx.x * 8 + wave;        // m*BWD_CH + ch
  int ch = unit % BWD_CH;
  int m = unit / BWD_CH;
  int hi = lane >> 4, lo = lane & 15;
  const float* Wm = W0 + (size_t)m * NX * 64;
  const float* g0 = ws + WS_G0 + m * 64;
  float mu = ws[WS_SUM] * (1.f / NX);
  float var = ws[WS_SUMSQ] * (1.f / NX) - mu * mu;
  float rs = rsqrtf(var + LN_EPS);
  float s1 = 0.f, s2 = 0.f;
  for (int tile = 0; tile < BWD_TILES; ++tile) {
    int i0 = (ch * BWD_TILES + tile) * 16;
    int row = i0 + lo;
    const float* wrow = Wm + (size_t)row * 64 + 2 * hi;
    v8f acc = {0.f,0.f,0.f,0.f,0.f,0.f,0.f,0.f};
    #pragma unroll
    for (int kt = 0; kt < 16; ++kt) {      // j0 = 4*kt
      v2f a, b;
      a.x = wrow[kt * 4 + 0];
      a.y = wrow[kt * 4 + 1];
      b.x = g0[kt * 4 + 2 * hi];
      b.y = g0[kt * 4 + 2 * hi + 1];
      acc = wmma4(a, b, acc);
    }
    if (lane == 0 || lane == 16) {
      int rbase = i0 + ((lane == 16) ? 8 : 0);
      #pragma unroll
      for (int v = 0; v < 8; ++v) {
        int i = rbase + v;
        float dxh = acc[v] * gamma[i];     // grad wrt xhat
        ws[WS_DXHAT + (size_t)m * NX + i] = dxh;
        s1 += dxh;
        float xh = (ws[WS_XRAW + i] - mu) * rs;
        s2 += dxh * xh;
      }
    }
  }
  if (lane == 0 || lane == 16) {
    atomicAdd(&ws[WS_S1 + m], s1);
    atomicAdd(&ws[WS_S2 + m], s2);
  }
}

// ---------------- K5: per-CV mean force + ensemble variance ----------------
__global__ void k_cvgrad(float* __restrict__ ws) {
  __shared__ float red[256];
  int tid = threadIdx.x;
  int c = blockIdx.x * 256 + tid;
  float ssl = 0.f;
  if (c < N_CVS) {
    float mu = ws[WS_SUM] * (1.f / NX);
    float var = ws[WS_SUMSQ] * (1.f / NX) - mu * mu;
    float rs = rsqrtf(var + LN_EPS);
    float cph = ws[WS_XRAW + c], sph = ws[WS_XRAW + N_CVS + c];
    float xh_c = (cph - mu) * rs, xh_s = (sph - mu) * rs;
    float f[N_MODELS];
    float mf = 0.f;
    #pragma unroll
    for (int m = 0; m < N_MODELS; ++m) {
      float m1 = ws[WS_S1 + m] * (1.f / NX);
      float m2 = ws[WS_S2 + m] * (1.f / NX);
      float d_c = ws[WS_DXHAT + (size_t)m * NX + c];
      float d_s = ws[WS_DXHAT + (size_t)m * NX + N_CVS + c];
      float dx_c = rs * (d_c - m1 - xh_c * m2);   // LN backward (jacrev-exact)
      float dx_s = rs * (d_s - m1 - xh_s * m2);
      float dEdcv = -sph * dx_c + cph * dx_s;     // through cos/sin
      f[m] = -dEdcv;                              // mean_forces = -jacrev
      mf += f[m];
    }
    mf *= (1.f / N_MODELS);
    ws[WS_MF + c] = mf;
    #pragma unroll
    for (int m = 0; m < N_MODELS; ++m) {
      float d = f[m] - mf;
      ssl += d * d;
    }
  }
  red[tid] = ssl;
  __syncthreads();
  for (int s = 128; s > 0; s >>= 1) {
    if (tid < s) red[tid] += red[tid + s];
    __syncthreads();
  }
  if (tid == 0) atomicAdd(&ws[WS_SS], red[0]);
}

// ---------------- K6: sigma + mean energy ----------------
__global__ void k_finalize(float* __restrict__ ws, float* __restrict__ out) {
  if (threadIdx.x == 0 && blockIdx.x == 0) {
    float SS = ws[WS_SS];
    float div = sqrtf(SS / (7.f * (float)N_CVS));  // ddof=1 over 8 models
    float isw = (3.f - div) / (3.f - 2.f);
    float fl = floorf(isw);
    float smooth = 0.5f * (1.f + cosf(PI_F * (1.f - isw)));
    float sigma = (fl > 0.f) ? 1.f : ((fl < 0.f) ? 0.f : smooth);
    ws[WS_SIGMA] = sigma;
    float e = 0.f;
    for (int m = 0; m < N_MODELS; ++m) e += ws[WS_E + m];
    out[0] = (e * (1.f / N_MODELS)) * sigma;
  }
}

// ---------------- K7: scatter forces ----------------
__global__ void k_scatter(const int* __restrict__ cidx,
                          const float* __restrict__ ws,
                          float* __restrict__ out) {
  int c = blockIdx.x * 256 + threadIdx.x;
  if (c >= N_CVS) return;
  float mfs = ws[WS_MF + c] * ws[WS_SIGMA];
  float* F = out + 1;
  const float* d = ws + WS_DCVS + (size_t)c * 12;
  #pragma unroll
  for (int a = 0; a < 4; ++a) {
    int pi = cidx[c * 4 + a];
    atomicAdd(&F[(size_t)pi * 3 + 0], mfs * d[a * 3 + 0]);
    atomicAdd(&F[(size_t)pi * 3 + 1], mfs * d[a * 3 + 1]);
    atomicAdd(&F[(size_t)pi * 3 + 2], mfs * d[a * 3 + 2]);
  }
}

extern "C" void kernel_launch(void* const* d_in, const int* in_sizes, int n_in,
                              void* d_out, int out_size, void* d_ws,
                              size_t ws_size, hipStream_t stream) {
  const float* positions = (const float*)d_in[0];
  const float* boxvectors = (const float*)d_in[1];
  const int* colvar_idx = (const int*)d_in[2];
  const float* gamma = (const float*)d_in[3];
  const float* beta = (const float*)d_in[4];
  const float* W0 = (const float*)d_in[5];
  const float* b0 = (const float*)d_in[6];
  const float* W1 = (const float*)d_in[7];
  const float* b1 = (const float*)d_in[8];
  const float* W2 = (const float*)d_in[9];
  const float* b2 = (const float*)d_in[10];
  const float* W3 = (const float*)d_in[11];
  const float* b3 = (const float*)d_in[12];
  const float* W4 = (const float*)d_in[13];
  const float* b4 = (const float*)d_in[14];
  float* out = (float*)d_out;
  float* ws = (float*)d_ws;

  // zero forces output (B128 bulk + scalar tail) + accumulator region
  int n4 = out_size >> 2;
  int ntail = out_size - (n4 << 2);
  k_zero4<<<4096, 256, 0, stream>>>((float4*)out, n4);
  if (ntail > 0)
    k_zero1<<<1, 32, 0, stream>>>(out + ((size_t)n4 << 2), ntail);
  k_zero4<<<2, 256, 0, stream>>>((float4*)(ws + WS_ACC_BEGIN),
                                 WS_ACC_COUNT / 4);
  // dihedrals + grads + LN statistics
  k_dihedral<<<(N_CVS + 255) / 256, 256, 0, stream>>>(positions, boxvectors,
                                                      colvar_idx, ws);
  // layer0 forward (streams 123 MB of W0 through WMMA); 12000 waves
  k_layer0_fwd<<<(N_MODELS * 4 * FWD_CH) / 8, 256, 0, stream>>>(W0, gamma,
                                                                beta, ws);
  // tiny hidden layers fwd+bwd
  k_mlp<<<N_MODELS, 64, 0, stream>>>(b0, W1, b1, W2, b2, W3, b3, W4, b4, ws);
  // layer0 backward (second W0 pass, L2-resident); 6000 waves
  k_layer0_bwd<<<(N_MODELS * BWD_CH) / 8, 256, 0, stream>>>(W0, gamma, ws);
  // per-CV mean forces + model divergence
  k_cvgrad<<<(N_CVS + 255) / 256, 256, 0, stream>>>(ws);
  // sigma + mean energy
  k_finalize<<<1, 32, 0, stream>>>(ws, out);
  // scatter into forces
  k_scatter<<<(N_CVS + 255) / 256, 256, 0, stream>>>(colvar_idx, ws, out);
}